// TransposedLocallyConnected2D_40183714021733
// MI455X (gfx1250) — compile-verified
//
#include <hip/hip_runtime.h>

// ---------------------------------------------------------------------------
// TransposedLocallyConnected2D on MI455X (gfx1250, wave32)
//   out[b,o,l] = sum_d unfold(x)[b,d,l] * W[d,l,o] + bias[o,l]
// One workgroup per spatial location l: a 64x576x128 f32 GEMM with
// V_WMMA_F32_16X16X4_F32. A (unfold) staged once in LDS; W streamed through a
// double-buffered LDS panel by the Tensor Data Mover (tensor_load_to_lds),
// synchronized with s_wait_tensorcnt + workgroup barrier.
// ---------------------------------------------------------------------------

typedef __attribute__((ext_vector_type(2))) float        v2f;
typedef __attribute__((ext_vector_type(8))) float        v8f;
typedef __attribute__((ext_vector_type(4))) unsigned int u32x4;
typedef __attribute__((ext_vector_type(8))) int          i32x8;
typedef __attribute__((ext_vector_type(4))) int          i32x4;

#define B_   64
#define C_   64
#define HW_  32
#define OC_  128
#define D_   576           // C*3*3
#define L_   1024          // 32*32
#define WROW (L_ * OC_)    // element stride between weight k-rows = 131072

#define AS_STRIDE 580      // 576 + 4  -> conflict-free ds_load_b64 A frags
#define WS_STRIDE 144      // 128 + 16 -> lane halves hit disjoint bank sets
#define KC        32       // k-chunk rows per LDS stage
#define NCHUNK    (D_ / KC)

// TDM: DMA one KC x 128 f32 weight tile (row stride WROW elements in memory)
// into LDS at byte offset lds_off, inserting 16 pad DWORDs after every 128
// DWORDs so the LDS image has row stride WS_STRIDE = 144 floats.
__device__ __forceinline__ void tdm_load_w_chunk(const float* gsrc,
                                                 unsigned lds_off) {
    const unsigned long long ga = (unsigned long long)(uintptr_t)gsrc;

    u32x4 g0;
    g0.x = 1u;                                   // count=1 (valid user D#)
    g0.y = lds_off;                              // LDS byte address
    g0.z = (unsigned)(ga & 0xffffffffu);         // global_addr[31:0]
    g0.w = (unsigned)((ga >> 32) & 0x01ffffffu)  // global_addr[56:32]
         | (2u << 30);                           // type = 2 ("image")

    i32x8 g1;
    g1[0] = (2 << 16)                            // data_size = 4 bytes
          | (1 << 20)                            // pad_enable
          | (6 << 22)                            // pad_interval: every 128 DW
          | (15 << 25);                          // pad_amount: 16 DWORDs
    g1[1] = (int)(128u << 16);                   // tensor_dim0 = 128 (lo16)
    g1[2] = (int)((unsigned)KC << 16);           // dim0 hi=0 | tensor_dim1=KC
    g1[3] = (int)(128u << 16);                   // dim1 hi=0 | tile_dim0=128
    g1[4] = KC;                                  // tile_dim1=KC | tile_dim2=0
    g1[5] = WROW;                                // tensor_dim0_stride lo32
    g1[6] = 0;                                   // stride hi | dim1_stride lo
    g1[7] = 0;                                   // dim1_stride hi (2-D tile)

    const i32x4 z4 = {0, 0, 0, 0};               // groups 2/3 unused (<=2D)
    const i32x8 z8 = {0, 0, 0, 0, 0, 0, 0, 0};   // extra group (clang-23 form)
    __builtin_amdgcn_tensor_load_to_lds(g0, g1, z4, z4, z8, /*cpol=*/0);
}

__global__ __launch_bounds__(256)
void tlc2d_wmma_f32_kernel(const float* __restrict__ x,
                           const float* __restrict__ w,
                           const float* __restrict__ bias,
                           float* __restrict__ out)
{
    extern __shared__ float smem[];
    float* As  = smem;                       // 64 x AS_STRIDE
    float* Ws0 = smem + B_ * AS_STRIDE;      // KC x WS_STRIDE
    float* Ws1 = Ws0 + KC * WS_STRIDE;       // KC x WS_STRIDE
    const unsigned ws0_off = (unsigned)(B_ * AS_STRIDE * sizeof(float));
    const unsigned ws1_off = ws0_off + (unsigned)(KC * WS_STRIDE * sizeof(float));

    const int l    = blockIdx.x;             // location index (oh*32+ow)
    const int oh   = l >> 5;
    const int ow   = l & 31;
    const int tid  = threadIdx.x;
    const int wave = tid >> 5;
    const int lane = tid & 31;
    const int hi   = lane >> 4;              // upper half-wave flag
    const int ln   = lane & 15;
    const int rt   = wave >> 1;              // row tile (0..3): batch rows rt*16..+15
    const int cg   = wave & 1;               // col group: 4 col tiles of 16 each

    const float* Wg = w + (size_t)l * OC_;   // k-row d lives at Wg + d*WROW

    // ---- TDM: start streaming weight chunk 0 into Ws0 ---------------------
    if (wave == 0)
        tdm_load_w_chunk(Wg, ws0_off);

    // ---- stage A = unfold(x)[:, :, l]  (64 x 576) into LDS, once ----------
    // d = c*9 + ki*3 + kj ; value = x[b, c, oh+ki-1, ow+kj-1] (zero padded)
    for (int e = tid; e < B_ * D_; e += 256) {     // 36864 = 144*256, uniform
        int b  = e / D_;
        int d  = e - b * D_;
        int c  = d / 9;
        int r9 = d - c * 9;
        int ki = r9 / 3;
        int kj = r9 - ki * 3;
        int h  = oh + ki - 1;
        int wc = ow + kj - 1;
        float v = 0.0f;
        if ((unsigned)h < 32u && (unsigned)wc < 32u)
            v = x[(((size_t)b * C_ + c) * HW_ + h) * HW_ + wc];
        As[b * AS_STRIDE + d] = v;
    }

    __builtin_amdgcn_s_wait_tensorcnt(0);    // chunk 0 resident (wave0; NOP else)
    __syncthreads();

    v8f acc[4] = {};                          // 4 col tiles of 16x16 f32

    for (int ch = 0; ch < NCHUNK; ++ch) {
        const float* Wbuf  = (ch & 1) ? Ws1 : Ws0;
        const int    kbase = ch * KC;
        const bool   more  = (ch + 1 < NCHUNK);

        // Kick the DMA for the next chunk before the WMMA burst (overlap).
        if (more && wave == 0)
            tdm_load_w_chunk(Wg + (size_t)(kbase + KC) * WROW,
                             (ch & 1) ? ws0_off : ws1_off);

        // ---- WMMA burst over this chunk: KC/4 = 8 k-steps ------------------
#pragma unroll
        for (int ks = 0; ks < KC / 4; ++ks) {
            const int k0 = ks * 4;
            // A frag (16x4): VGPR0/1 hold K0,K1 (lanes 0-15) / K2,K3 (16-31)
            v2f afrag = *(const v2f*)&As[(rt * 16 + ln) * AS_STRIDE
                                         + kbase + k0 + 2 * hi];
#pragma unroll
            for (int u = 0; u < 4; ++u) {
                const int n  = (cg * 4 + u) * 16 + ln;   // output channel
                const int lr = k0 + hi;                  // B rows K=lr, lr+2
                v2f bfrag;
                bfrag.x = Wbuf[lr * WS_STRIDE + n];
                bfrag.y = Wbuf[(lr + 2) * WS_STRIDE + n];
                acc[u] = __builtin_amdgcn_wmma_f32_16x16x4_f32(
                    /*neg_a=*/false, afrag, /*neg_b=*/false, bfrag,
                    /*c_mod=*/(short)0, acc[u],
                    /*reuse_a=*/false, /*reuse_b=*/false);
            }
        }

        if (more) {
            __builtin_amdgcn_s_wait_tensorcnt(0);  // next chunk landed (wave0)
            __syncthreads();                       // publish to all waves
        }
    }

    // ---- epilogue: bias add + store (lanes 0-15 store 4 KB apart, stride L_) ------
#pragma unroll
    for (int u = 0; u < 4; ++u) {
        const int o  = (cg * 4 + u) * 16 + ln;
        const float bv = bias[(size_t)o * L_ + l];
#pragma unroll
        for (int v = 0; v < 8; ++v) {
            const int br = rt * 16 + hi * 8 + v;   // C/D layout: M=v (+8 hi)
            out[((size_t)br * OC_ + o) * L_ + l] = acc[u][v] + bv;
        }
    }
}

extern "C" void kernel_launch(void* const* d_in, const int* in_sizes, int n_in,
                              void* d_out, int out_size, void* d_ws, size_t ws_size,
                              hipStream_t stream) {
    (void)in_sizes; (void)n_in; (void)d_ws; (void)ws_size; (void)out_size;
    const float* x    = (const float*)d_in[0];
    const float* w    = (const float*)d_in[1];
    const float* bias = (const float*)d_in[2];
    float* out = (float*)d_out;

    const size_t shmem =
        (size_t)(B_ * AS_STRIDE + 2 * KC * WS_STRIDE) * sizeof(float); // ~185 KB

    static bool attr_set = false; // attribute config is idempotent/deterministic
    if (!attr_set) {
        (void)hipFuncSetAttribute((const void*)tlc2d_wmma_f32_kernel,
                                  hipFuncAttributeMaxDynamicSharedMemorySize,
                                  (int)shmem);
        attr_set = true;
    }

    tlc2d_wmma_f32_kernel<<<dim3(L_), dim3(256), shmem, stream>>>(x, w, bias, out);
}